// TestModule_73005854097868
// MI455X (gfx1250) — compile-verified
//
#include <hip/hip_runtime.h>

// ---------------------------------------------------------------------------
// GraphConv x2 + log_softmax for MI455X (gfx1250).
//   - transform-then-aggregate: segment_sum(x[s]) @ W.T == segment_sum((x@W.T)[s])
//     => edge traffic 16 floats/edge (layer1), 8 floats/edge (layer2) vs 64.
//   - GEMMs on V_WMMA_F32_16X16X4_F32 (full fp32), one wave per 16-node tile,
//     two interleaved accumulator chains (rel + root) for ILP.
//   - GEMM kernels seed the aggregation buffers with root-term + bias, so no
//     zero-fill passes and one fewer read stream in the combines.
//   - scatter kernels stage edge indices in LDS (one int64 load per edge,
//     broadcast to 16/8 feature lanes) -> 16x/8x less index traffic.
// ---------------------------------------------------------------------------

typedef float v2f __attribute__((ext_vector_type(2)));
typedef float v8f __attribute__((ext_vector_type(8)));

#define N_NODES   100000
#define N_EDGESC  1600000
#define IN_F      64
#define HID       16
#define NCLS      7
#define NCLS_PAD  8
#define NTILES    (N_NODES / 16)      // 6250, exact
#define EPB       256                 // edges per scatter block (E % 256 == 0)

// ---------------------------------------------------------------------------
// Layer-1: t_rel = x @ W1_rel.T ; agg1 = x @ W1_root.T + b1   (both [N,16])
// ---------------------------------------------------------------------------
__global__ __launch_bounds__(128)
void gemm1_kernel(const float* __restrict__ x,        // [N][64]
                  const float* __restrict__ W1_rel,   // [16][64]
                  const float* __restrict__ W1_root,  // [16][64]
                  const float* __restrict__ b1,       // [16]
                  float* __restrict__ t_rel,          // [N][16]
                  float* __restrict__ agg1) {         // [N][16] (seeded)
  const int lane = threadIdx.x & 31;
  const int wave = threadIdx.x >> 5;
  int tile = blockIdx.x * 4 + wave;
  if (tile >= NTILES) tile = NTILES - 1;  // clamp: EXEC all-ones at every WMMA

  const int mr = lane & 15;          // A row / B,D column
  const int kh = (lane >> 4) * 2;    // K sub-offset within the 4-wide K tile

  const float* xrow = x + (size_t)(tile * 16 + mr) * IN_F + kh;
  const float* wrl  = W1_rel  + mr * IN_F + kh;
  const float* wrt  = W1_root + mr * IN_F + kh;
  const float  bias = b1[mr];        // bias for output column n = mr

  v8f acc_rel  = {0.f,0.f,0.f,0.f,0.f,0.f,0.f,0.f};
  v8f acc_root = {0.f,0.f,0.f,0.f,0.f,0.f,0.f,0.f};

#pragma unroll
  for (int kb = 0; kb < IN_F; kb += 4) {
    v2f a    = *(const v2f*)(xrow + kb);                    // A[m][kb+kh..+1]
    v2f brel = *(const v2f*)(wrl + kb);                     // B[kb+kh..+1][n]
    v2f brt  = *(const v2f*)(wrt + kb);
    acc_rel  = __builtin_amdgcn_wmma_f32_16x16x4_f32(
        false, a, false, brel, (short)0, acc_rel, false, false);
    acc_root = __builtin_amdgcn_wmma_f32_16x16x4_f32(
        false, a, false, brt,  (short)0, acc_root, false, false);
  }

  // D layout: lane holds column n=mr, rows m = v + 8*(lane>>4)
  const int mbase = tile * 16 + (lane >> 4) * 8;
#pragma unroll
  for (int v = 0; v < 8; ++v) {
    t_rel[(size_t)(mbase + v) * HID + mr] = acc_rel[v];
    agg1 [(size_t)(mbase + v) * HID + mr] = acc_root[v] + bias;
  }
}

// ---------------------------------------------------------------------------
// Layer-2: u_rel = h @ W2_rel.T ; agg2 = h @ W2_root.T + b2  ([N,8], col7=0)
// ---------------------------------------------------------------------------
__global__ __launch_bounds__(128)
void gemm2_kernel(const float* __restrict__ h,        // [N][16]
                  const float* __restrict__ W2_rel,   // [7][16]
                  const float* __restrict__ W2_root,  // [7][16]
                  const float* __restrict__ b2,       // [7]
                  float* __restrict__ u_rel,          // [N][8]
                  float* __restrict__ agg2) {         // [N][8] (seeded)
  const int lane = threadIdx.x & 31;
  const int wave = threadIdx.x >> 5;
  int tile = blockIdx.x * 4 + wave;
  if (tile >= NTILES) tile = NTILES - 1;

  const int  mr = lane & 15;
  const int  kh = (lane >> 4) * 2;
  const bool valid_col = (mr < NCLS);
  const int  wrow = valid_col ? mr : 0;   // branchless pad: select, not diverge

  const float* hrow = h + (size_t)(tile * 16 + mr) * HID + kh;
  const float* wrl  = W2_rel  + wrow * HID + kh;
  const float* wrt  = W2_root + wrow * HID + kh;
  const float  bias = valid_col ? b2[mr] : 0.f;

  v8f acc_rel  = {0.f,0.f,0.f,0.f,0.f,0.f,0.f,0.f};
  v8f acc_root = {0.f,0.f,0.f,0.f,0.f,0.f,0.f,0.f};

#pragma unroll
  for (int kb = 0; kb < HID; kb += 4) {
    v2f a    = *(const v2f*)(hrow + kb);
    v2f brel = *(const v2f*)(wrl + kb);
    v2f brt  = *(const v2f*)(wrt + kb);
    if (!valid_col) { brel.x = 0.f; brel.y = 0.f; brt.x = 0.f; brt.y = 0.f; }
    acc_rel  = __builtin_amdgcn_wmma_f32_16x16x4_f32(
        false, a, false, brel, (short)0, acc_rel, false, false);
    acc_root = __builtin_amdgcn_wmma_f32_16x16x4_f32(
        false, a, false, brt,  (short)0, acc_root, false, false);
  }

  const int mbase = tile * 16 + (lane >> 4) * 8;
  if (mr < NCLS_PAD) {   // divergence only AFTER all WMMAs — legal
#pragma unroll
    for (int v = 0; v < 8; ++v) {
      u_rel[(size_t)(mbase + v) * NCLS_PAD + mr] = acc_rel[v];
      agg2 [(size_t)(mbase + v) * NCLS_PAD + mr] = acc_root[v] + bias;
    }
  }
}

// ---------------------------------------------------------------------------
// Edge scatter-adds with LDS-staged indices.
// Block = 256 threads = 256 edges. Phase 1: one int64 src + dst load per edge
// into LDS. Phase 2: (edge, feature) pairs; LDS index reads are 16-lane (or
// 8-lane) broadcasts, payload loads coalesce, atomics go to L2.
// ---------------------------------------------------------------------------
__global__ __launch_bounds__(EPB)
void scatter16_kernel(const long long* __restrict__ src,
                      const long long* __restrict__ dst,
                      const float* __restrict__ t_rel,
                      float* __restrict__ agg1) {
  __shared__ int sS[EPB];
  __shared__ int sD[EPB];
  const int tid = threadIdx.x;
  const int e   = blockIdx.x * EPB + tid;      // E % EPB == 0, no guard
  sS[tid] = (int)src[e];
  sD[tid] = (int)dst[e];
  __syncthreads();
#pragma unroll
  for (int it = 0; it < 16; ++it) {            // 256 edges * 16 feats / 256 thr
    const int p  = it * EPB + tid;
    const int el = p >> 4;                     // local edge 0..255
    const int f  = p & 15;                     // feature 0..15
    const int s = sS[el];
    const int d = sD[el];
    atomicAdd(&agg1[(size_t)d * HID + f], t_rel[(size_t)s * HID + f]);
  }
}

__global__ __launch_bounds__(EPB)
void scatter8_kernel(const long long* __restrict__ src,
                     const long long* __restrict__ dst,
                     const float* __restrict__ u_rel,
                     float* __restrict__ agg2) {
  __shared__ int sS[EPB];
  __shared__ int sD[EPB];
  const int tid = threadIdx.x;
  const int e   = blockIdx.x * EPB + tid;
  sS[tid] = (int)src[e];
  sD[tid] = (int)dst[e];
  __syncthreads();
#pragma unroll
  for (int it = 0; it < 8; ++it) {             // 256 edges * 8 feats / 256 thr
    const int p  = it * EPB + tid;
    const int el = p >> 3;
    const int f  = p & 7;
    const int s = sS[el];
    const int d = sD[el];
    atomicAdd(&agg2[(size_t)d * NCLS_PAD + f], u_rel[(size_t)s * NCLS_PAD + f]);
  }
}

// ---------------------------------------------------------------------------
// h = relu(agg1), vectorized float4 (N*16 elements, divisible by 4)
// ---------------------------------------------------------------------------
__global__ void relu4_kernel(const float4* __restrict__ in,
                             float4* __restrict__ out, int n4) {
  int i = blockIdx.x * blockDim.x + threadIdx.x;
  if (i < n4) {
    float4 v = in[i];
    v.x = v.x > 0.f ? v.x : 0.f;
    v.y = v.y > 0.f ? v.y : 0.f;
    v.z = v.z > 0.f ? v.z : 0.f;
    v.w = v.w > 0.f ? v.w : 0.f;
    out[i] = v;
  }
}

// ---------------------------------------------------------------------------
// out = log_softmax(agg2[:, :7])  (agg2 already holds full logits)
// ---------------------------------------------------------------------------
__global__ void final_kernel(const float* __restrict__ agg2,
                             float* __restrict__ out) {
  int n = blockIdx.x * blockDim.x + threadIdx.x;
  if (n >= N_NODES) return;
  float l[NCLS];
  float m = -3.402823466e+38f;
#pragma unroll
  for (int c = 0; c < NCLS; ++c) {
    l[c] = agg2[(size_t)n * NCLS_PAD + c];
    m = fmaxf(m, l[c]);
  }
  float s = 0.f;
#pragma unroll
  for (int c = 0; c < NCLS; ++c) s += expf(l[c] - m);
  float lse = logf(s);
#pragma unroll
  for (int c = 0; c < NCLS; ++c) out[(size_t)n * NCLS + c] = l[c] - m - lse;
}

// ---------------------------------------------------------------------------
// Launcher
// ---------------------------------------------------------------------------
extern "C" void kernel_launch(void* const* d_in, const int* in_sizes, int n_in,
                              void* d_out, int out_size, void* d_ws, size_t ws_size,
                              hipStream_t stream) {
  const float*     x       = (const float*)d_in[0];
  const long long* ei      = (const long long*)d_in[1];  // int64 [2][E]
  const float*     W1_rel  = (const float*)d_in[2];
  const float*     b1_rel  = (const float*)d_in[3];
  const float*     W1_root = (const float*)d_in[4];
  const float*     W2_rel  = (const float*)d_in[5];
  const float*     b2_rel  = (const float*)d_in[6];
  const float*     W2_root = (const float*)d_in[7];
  float* out = (float*)d_out;

  const long long* src = ei;
  const long long* dst = ei + N_EDGESC;

  // Workspace (floats): 3*[N,16] + 2*[N,8] = 25.6 MB; fully rewritten per call
  float* ws    = (float*)d_ws;
  float* t_rel = ws;                               // [N][16]
  float* agg1  = t_rel + (size_t)N_NODES * HID;    // [N][16] seeded by gemm1
  float* h     = agg1  + (size_t)N_NODES * HID;    // [N][16]
  float* u_rel = h     + (size_t)N_NODES * HID;    // [N][8]
  float* agg2  = u_rel + (size_t)N_NODES * NCLS_PAD; // [N][8] seeded by gemm2

  const int B = 256;

  // 1. layer-1 transforms (WMMA); agg1 <- x@W1_root.T + b1
  gemm1_kernel<<<(NTILES + 3) / 4, 128, 0, stream>>>(
      x, W1_rel, W1_root, b1_rel, t_rel, agg1);

  // 2. layer-1 aggregation: agg1 += segment_sum(t_rel[src])
  scatter16_kernel<<<N_EDGESC / EPB, EPB, 0, stream>>>(src, dst, t_rel, agg1);

  // 3. h = relu(agg1)
  {
    int n4 = N_NODES * HID / 4;
    relu4_kernel<<<(n4 + B - 1) / B, B, 0, stream>>>(
        (const float4*)agg1, (float4*)h, n4);
  }

  // 4. layer-2 transforms (WMMA); agg2 <- h@W2_root.T + b2
  gemm2_kernel<<<(NTILES + 3) / 4, 128, 0, stream>>>(
      h, W2_rel, W2_root, b2_rel, u_rel, agg2);

  // 5. layer-2 aggregation: agg2 += segment_sum(u_rel[src])
  scatter8_kernel<<<N_EDGESC / EPB, EPB, 0, stream>>>(src, dst, u_rel, agg2);

  // 6. log_softmax over 7 classes
  final_kernel<<<(N_NODES + B - 1) / B, B, 0, stream>>>(agg2, out);
}